// NRBacksolveFunction_35725537968822
// MI455X (gfx1250) — compile-verified
//
#include <hip/hip_runtime.h>
#include <hip/hip_bf16.h>

// Problem constants (from reference setup_inputs)
#define RHO   365
#define BS    8192
#define NY    8
#define NF    4
#define NFLUX 4
#define NITER 5
#define BLOCK 32   // ONE wave32 per workgroup: s_barrier is a HW NOP (ISA §3.1),
                   // so __syncthreads() is a free compiler fence; no cross-wave coupling.
#define NWB   16   // merged B width: cols 0..7 = W1, 8..11 = Wf, 12..15 = 0

typedef float v2f __attribute__((ext_vector_type(2)));
typedef float v8f __attribute__((ext_vector_type(8)));

// Full-precision CDNA5 f32 WMMA: D(16x16) = A(16x4) * B(4x16) + C
__device__ __forceinline__ v8f wmma4(v2f a, v2f b, v8f c) {
  return __builtin_amdgcn_wmma_f32_16x16x4_f32(
      /*neg_a=*/false, a, /*neg_b=*/false, b,
      /*c_mod=*/(short)0, c, /*reuse_a=*/false, /*reuse_b=*/false);
}

// Wave-cooperative matvec: per-lane row yv[8] (lane = batch element) times the
// preloaded merged B operand (8 x 16: [W1 | Wf | 0]) -> per-lane zout[nout]
// (nout = 8 for Newton z, 12 for merged z+flux). Two 16-row WMMA tiles cover
// the wave's 32 rows; LDS staging performs the layout transposes.
__device__ __forceinline__ void wave_matvec(const float yv[NY], float* zout, int nout,
                                            float* ys, float* zs,
                                            v2f b0, v2f bk1, int ln)
{
  const int mr = ln & 15;     // row within tile (A) / column (C/D)
  const int hi = ln >> 4;     // lane half selects K pair / M half
  const int k0 = hi ? 2 : 0;
  const int k1 = hi ? 3 : 1;

  __syncthreads();                       // free fence (single-wave WG)
  #pragma unroll
  for (int j = 0; j < NY; ++j) ys[ln * NY + j] = yv[j];
  __syncthreads();

  // A operand (16x4 f32): lanes 0-15 hold K=0,1; lanes 16-31 hold K=2,3.
  v2f a00, a01, a10, a11;
  a00.x = ys[mr * NY + k0];             a00.y = ys[mr * NY + k1];
  a01.x = ys[mr * NY + 4 + k0];         a01.y = ys[mr * NY + 4 + k1];
  a10.x = ys[(16 + mr) * NY + k0];      a10.y = ys[(16 + mr) * NY + k1];
  a11.x = ys[(16 + mr) * NY + 4 + k0];  a11.y = ys[(16 + mr) * NY + 4 + k1];

  v8f c0 = {};
  c0 = wmma4(a00, b0, c0);   // K = 0..3
  c0 = wmma4(a01, bk1, c0);  // K = 4..7
  v8f c1 = {};
  c1 = wmma4(a10, b0, c1);
  c1 = wmma4(a11, bk1, c1);

  __syncthreads();
  // Transpose C/D layout (M = r + 8*hi, N = mr) back to per-lane rows.
  if (mr < nout) {
    #pragma unroll
    for (int r = 0; r < 8; ++r) {
      zs[(hi * 8 + r) * NWB + mr]      = c0[r];
      zs[(16 + hi * 8 + r) * NWB + mr] = c1[r];
    }
  }
  __syncthreads();
  #pragma unroll
  for (int j = 0; j < 12; ++j)
    if (j < nout) zout[j] = zs[ln * NWB + j];
}

__global__ void __launch_bounds__(BLOCK)
nr_kernel(const float* __restrict__ y0, const float* __restrict__ dtp,
          const float* __restrict__ inp, const float* __restrict__ W1,
          const float* __restrict__ Wx, const float* __restrict__ b1,
          const float* __restrict__ Wf, const float* __restrict__ alphap,
          float* __restrict__ outY, float* __restrict__ outFlux,
          float* __restrict__ outRes, unsigned* __restrict__ wsMax)
{
  __shared__ float shWB[NY * NWB];   // merged [W1 | Wf | 0], row-major (k-major)
  __shared__ float shWx[NF * NY];
  __shared__ float shB1[NY];
  __shared__ float stA[32 * NY];     // wave A staging
  __shared__ float stZ[32 * NWB];    // wave result staging

  const int ln = threadIdx.x;                 // lane id (single wave)
  const int b  = blockIdx.x * BLOCK + ln;     // batch element

  // Stage weights: B = [W1 (8 cols) | Wf (4 cols) | zeros (4 cols)]
  #pragma unroll
  for (int i = ln; i < NY * NWB; i += BLOCK) {
    const int k = i >> 4, n = i & 15;
    float v = 0.0f;
    if (n < NY)                 v = W1[k * NY + n];
    else if (n < NY + NFLUX)    v = Wf[k * NFLUX + (n - NY)];
    shWB[i] = v;
  }
  if (ln < NF * NY) shWx[ln] = Wx[ln];
  if (ln < NY)      shB1[ln] = b1[ln];
  __syncthreads();

  const float dt    = dtp[0];
  const float alpha = alphap[0];
  const float cd    = 1.0f + dt * alpha;      // Jacobian diagonal shift

  // Loop-invariant B operands (4x16 f32 layout: VGPR0=K{0|2}, VGPR1=K{1|3}).
  v2f Bw0, Bw1;
  {
    const int mr = ln & 15, hi = ln >> 4;
    const int k0 = hi ? 2 : 0, k1 = hi ? 3 : 1;
    Bw0.x = shWB[(k0)     * NWB + mr];  Bw0.y = shWB[(k1)     * NWB + mr];
    Bw1.x = shWB[(k0 + 4) * NWB + mr];  Bw1.y = shWB[(k1 + 4) * NWB + mr];
  }

  float yold[NY];
  {
    const float4 a = ((const float4*)(y0 + (size_t)b * NY))[0];
    const float4 c = ((const float4*)(y0 + (size_t)b * NY))[1];
    yold[0] = a.x; yold[1] = a.y; yold[2] = a.z; yold[3] = a.w;
    yold[4] = c.x; yold[5] = c.y; yold[6] = c.z; yold[7] = c.w;
  }

  for (int t = 0; t < RHO; ++t) {
    const size_t rowb = (size_t)t * BS + b;

    const float4 x4 = *(const float4*)(inp + rowb * NF);
    if (t + 1 < RHO)                                     // hide next step's load
      __builtin_prefetch(inp + (rowb + (size_t)BS) * NF, 0, 0);

    // per-step affine part: zb = b1 + x_t @ Wx
    float zb[NY];
    #pragma unroll
    for (int j = 0; j < NY; ++j)
      zb[j] = shB1[j] + x4.x * shWx[0 * NY + j] + x4.y * shWx[1 * NY + j]
                      + x4.z * shWx[2 * NY + j] + x4.w * shWx[3 * NY + j];

    float yv[NY];
    #pragma unroll
    for (int j = 0; j < NY; ++j) yv[j] = yold[j];

    // ---- Newton iterations: F(y) = y - yold - dt*(tanh(z) - alpha*y) ----
    for (int it = 0; it < NITER; ++it) {
      float zlin[NY];
      wave_matvec(yv, zlin, NY, stA, stZ, Bw0, Bw1, ln);   // y @ W1 (WMMA)

      float Fv[NY], s2[NY];
      #pragma unroll
      for (int j = 0; j < NY; ++j) {
        const float z  = zlin[j] + zb[j];
        const float tj = tanhf(z);
        s2[j] = 1.0f - tj * tj;
        const float f = tj - alpha * yv[j];
        Fv[j] = yv[j] - yold[j] - dt * f;
      }

      // J[k][j] = (1+dt*alpha)*(k==j) - dt * s2[k] * W1[j][k]
      float Jm[NY][NY];
      #pragma unroll
      for (int k = 0; k < NY; ++k) {
        #pragma unroll
        for (int j = 0; j < NY; ++j)
          Jm[k][j] = -dt * s2[k] * shWB[j * NWB + k];      // W1[j][k]
        Jm[k][k] += cd;
      }

      // 8x8 Gaussian elimination, fully unrolled (J is diagonally dominant)
      #pragma unroll
      for (int p = 0; p < NY; ++p) {
        const float inv = 1.0f / Jm[p][p];
        #pragma unroll
        for (int r = p + 1; r < NY; ++r) {
          const float fac = Jm[r][p] * inv;
          #pragma unroll
          for (int c = p + 1; c < NY; ++c) Jm[r][c] -= fac * Jm[p][c];
          Fv[r] -= fac * Fv[p];
        }
      }
      float sol[NY];
      #pragma unroll
      for (int p = NY - 1; p >= 0; --p) {
        float acc = Fv[p];
        #pragma unroll
        for (int c = p + 1; c < NY; ++c) acc -= Jm[p][c] * sol[c];
        sol[p] = acc / Jm[p][p];
      }
      #pragma unroll
      for (int j = 0; j < NY; ++j) yv[j] -= sol[j];
    }

    // ---- merged final pass: one WMMA matvec gives z(ynew) AND ynew@Wf ----
    float zfx[12];
    wave_matvec(yv, zfx, 12, stA, stZ, Bw0, Bw1, ln);

    float Fv[NY], dy[NY];
    float mloc = 0.0f;
    #pragma unroll
    for (int j = 0; j < NY; ++j) {
      const float z = zfx[j] + zb[j];
      const float f = tanhf(z) - alpha * yv[j];
      Fv[j] = yv[j] - yold[j] - dt * f;
      dy[j] = f;
      mloc = fmaxf(mloc, fabsf(Fv[j]));
    }

    float ysol[NY];
    #pragma unroll
    for (int j = 0; j < NY; ++j) ysol[j] = yold[j] + dt * dy[j];

    // vectorized stores
    ((float4*)(outY + rowb * NY))[0] = make_float4(ysol[0], ysol[1], ysol[2], ysol[3]);
    ((float4*)(outY + rowb * NY))[1] = make_float4(ysol[4], ysol[5], ysol[6], ysol[7]);
    ((float4*)(outRes + rowb * NY))[0] = make_float4(Fv[0], Fv[1], Fv[2], Fv[3]);
    ((float4*)(outRes + rowb * NY))[1] = make_float4(Fv[4], Fv[5], Fv[6], Fv[7]);
    *(float4*)(outFlux + rowb * NFLUX) =
        make_float4(fmaxf(zfx[8], 0.f) * dt, fmaxf(zfx[9], 0.f) * dt,
                    fmaxf(zfx[10], 0.f) * dt, fmaxf(zfx[11], 0.f) * dt);

    // wave max|F| via shuffle butterfly, then one global atomic per wave.
    // Non-negative float bits are u32-monotone.
    #pragma unroll
    for (int off = 16; off > 0; off >>= 1)
      mloc = fmaxf(mloc, __shfl_xor(mloc, off, 32));
    if (ln == 0) atomicMax(wsMax + t, __float_as_uint(mloc));

    #pragma unroll
    for (int j = 0; j < NY; ++j) yold[j] = ysol[j];   // scan carry = ysol
  }
}

__global__ void ws_init(unsigned* __restrict__ ws) {
  const int i = blockIdx.x * blockDim.x + threadIdx.x;
  if (i < RHO) ws[i] = 0u;
}

__global__ void flag_final(const unsigned* __restrict__ ws,
                           float* __restrict__ outFlag) {
  const int i = blockIdx.x * blockDim.x + threadIdx.x;
  if (i < RHO) outFlag[i] = (__uint_as_float(ws[i]) < 1e-6f) ? 1.0f : 0.0f;
}

extern "C" void kernel_launch(void* const* d_in, const int* in_sizes, int n_in,
                              void* d_out, int out_size, void* d_ws, size_t ws_size,
                              hipStream_t stream) {
  // setup_inputs order: y0, delta_t, inp, W1, Wx, b1, Wf, alpha, nflux
  const float* y0     = (const float*)d_in[0];
  const float* dtp    = (const float*)d_in[1];
  const float* inp    = (const float*)d_in[2];
  const float* W1     = (const float*)d_in[3];
  const float* Wx     = (const float*)d_in[4];
  const float* b1     = (const float*)d_in[5];
  const float* Wf     = (const float*)d_in[6];
  const float* alphap = (const float*)d_in[7];
  // d_in[8] = nflux (compile-time NFLUX here)

  float* outY    = (float*)d_out;                              // [RHO,BS,NY]
  float* outFlux = outY + (size_t)RHO * BS * NY;               // [RHO,BS,NFLUX]
  float* outRes  = outFlux + (size_t)RHO * BS * NFLUX;         // [RHO,BS,NY]
  float* outFlag = outRes + (size_t)RHO * BS * NY;             // [RHO]

  unsigned* wsMax = (unsigned*)d_ws;   // RHO u32 accumulators (needs 1460 B)

  ws_init<<<(RHO + 255) / 256, 256, 0, stream>>>(wsMax);
  nr_kernel<<<BS / BLOCK, BLOCK, 0, stream>>>(y0, dtp, inp, W1, Wx, b1, Wf,
                                              alphap, outY, outFlux, outRes,
                                              wsMax);
  flag_final<<<(RHO + 255) / 256, 256, 0, stream>>>(wsMax, outFlag);
}